// PointChain_14242111554181
// MI455X (gfx1250) — compile-verified
//
#include <hip/hip_runtime.h>
#include <hip/hip_bf16.h>
#include <math.h>

// ---------------- problem constants ----------------
#define NFRAG   1024
#define FRAGLEN 512
#define NPTS    (NFRAG * FRAGLEN)   // 524288
#define DY      128
#define DISTC   3.8f
#define NITER   10
#define CONV    1e-4f

// ---------------- workspace layout (floats) ----------------
#define C_M      3    // M[3][3] row-major: consts[3 + j*3 + c]
#define C_V      12   // v[3]
#define C_ALPHA  15
#define C_CNORM  16
#define C_CT     17   // ct[11]
#define C_AAPPLY 28
#define C_DONE   29
#define C_ASUM   30
#define WUE_OFF  64                      // Wu_eff[128][3]
#define RIA_OFF  448
#define N3       (NPTS * 3)              // 1572864
#define LAM_OFF  (RIA_OFF + N3)
#define L_OFF    (LAM_OFF + N3)

typedef float v2f __attribute__((ext_vector_type(2)));
typedef float v8f __attribute__((ext_vector_type(8)));

__device__ inline float wave_reduce_add(float v) {
  #pragma unroll
  for (int o = 16; o > 0; o >>= 1) v += __shfl_down(v, o, 32);
  return v;
}

// ---------------- K0: build Wu_eff, M, v; init scalars ----------------
__global__ __launch_bounds__(128) void kprep(const float* __restrict__ Wp,
                                             const float* __restrict__ Wu,
                                             const float* __restrict__ bu,
                                             float* ws) {
  __shared__ float wue_s[DY * 3];
  int t = threadIdx.x;                       // 0..127, one per output feature d
  #pragma unroll
  for (int j = 0; j < 3; ++j) {
    float w = Wu[t * 18 + j] + Wu[t * 18 + j + 3] + Wu[t * 18 + j + 6];
    wue_s[t * 3 + j] = w;
    ws[WUE_OFF + t * 3 + j] = w;
  }
  __syncthreads();
  if (t < 9) {                               // M[j][c] = sum_d Wu_eff[d][j] * Wp[c][d]
    int j = t / 3, c = t % 3;
    float acc = 0.f;
    for (int d = 0; d < DY; ++d) acc += wue_s[d * 3 + j] * Wp[c * DY + d];
    ws[C_M + j * 3 + c] = acc;
  } else if (t < 12) {                       // v[c] = sum_d b_u[d] * Wp[c][d]
    int c = t - 9;
    float acc = 0.f;
    for (int d = 0; d < DY; ++d) acc += bu[d] * Wp[c * DY + d];
    ws[C_V + c] = acc;
  } else if (t == 12) {
    ws[C_ALPHA] = 1.0f; ws[C_CNORM] = 0.f; ws[C_AAPPLY] = 0.f;
    ws[C_DONE] = 0.f;   ws[C_ASUM]  = 0.f;
    for (int k = 0; k < 11; ++k) ws[C_CT + k] = 0.f;
  }
}

// ---------------- K_project: ria = y @ Wp3.T + bp3 via V_WMMA_F32_16X16X4_F32
// Per wave: D(16x16) = A(Wp rows 0..15, 16x4 K-chunk) x B(y.T 4x16 points),
// accumulated over K=128. Rows 0..2 of D are the 3 projected coords.
__global__ __launch_bounds__(256) void kproj(const float* __restrict__ y,
                                             const float* __restrict__ Wp,
                                             const float* __restrict__ bp,
                                             float* ws) {
  const int lane = threadIdx.x & 31;
  const int wave = threadIdx.x >> 5;
  const int tile = blockIdx.x * 8 + wave;    // 16 points per wave
  const int pb   = tile * 16;
  const int half = lane >> 4;                // ISA: lanes 16-31 hold K+2
  const int l16  = lane & 15;

  const float* wpRow = Wp + (size_t)l16 * DY;          // A row = l16
  const float* yRow  = y  + (size_t)(pb + l16) * DY;   // B col (point) = l16
  v8f c = {};
  #pragma unroll
  for (int kb = 0; kb < DY; kb += 4) {
    if ((kb & 31) == 0) __builtin_prefetch(yRow + kb + 64, 0, 0);
    const int k0 = kb + half * 2;
    v2f a, b;
    a[0] = wpRow[k0]; a[1] = wpRow[k0 + 1];
    b[0] = yRow[k0];  b[1] = yRow[k0 + 1];
    c = __builtin_amdgcn_wmma_f32_16x16x4_f32(false, a, false, b,
                                              (short)0, c, false, false);
  }
  // D layout: VGPR r = row M=r on lanes 0-15 (N = lane). Features 0..2 -> c[0..2].
  if (half == 0) {
    const size_t p = (size_t)(pb + l16);
    float* ria = ws + RIA_OFF;
    float* L   = ws + L_OFF;
    #pragma unroll
    for (int j = 0; j < 3; ++j) {
      ria[p * 3 + j] = c[j] + bp[j];
      L[p * 3 + j]   = 0.f;
    }
  }
}

// ---------------- KA: lam3 per point + baseline cnorm ----------------
__global__ __launch_bounds__(256) void kA(float* ws) {
  const int n = blockIdx.x * 256 + threadIdx.x;        // grid covers NPTS exactly
  const int l = n & (FRAGLEN - 1);
  const float* ria = ws + RIA_OFF;
  float* lam = ws + LAM_OFF;

  float rl[3];
  #pragma unroll
  for (int j = 0; j < 3; ++j) rl[j] = ria[(size_t)n * 3 + j];

  float sR[3] = {0.f, 0.f, 0.f}, sL[3] = {0.f, 0.f, 0.f};
  float c2 = 0.f;
  if (l < FRAGLEN - 1) {                               // bond (l, l+1)
    float dx[3], d2 = 0.f;
    #pragma unroll
    for (int j = 0; j < 3; ++j) { dx[j] = ria[(size_t)(n + 1) * 3 + j] - rl[j]; d2 += dx[j] * dx[j]; }
    float inv = rsqrtf(d2);
    float cc  = DISTC * inv - 1.f;
    float cf  = DISTC * inv * inv * inv * cc;          // DIST * d2^{-1.5} * c
    #pragma unroll
    for (int j = 0; j < 3; ++j) sR[j] = cf * dx[j];
    c2 = cc * cc;
  }
  if (l > 0) {                                          // bond (l-1, l)
    float dx[3], d2 = 0.f;
    #pragma unroll
    for (int j = 0; j < 3; ++j) { dx[j] = rl[j] - ria[(size_t)(n - 1) * 3 + j]; d2 += dx[j] * dx[j]; }
    float inv = rsqrtf(d2);
    float cc  = DISTC * inv - 1.f;
    float cf  = DISTC * inv * inv * inv * cc;
    #pragma unroll
    for (int j = 0; j < 3; ++j) sL[j] = cf * dx[j];
  }
  #pragma unroll
  for (int j = 0; j < 3; ++j) lam[(size_t)n * 3 + j] = sR[j] - sL[j];   // diffT scatter

  // hierarchical reduction of sum(c^2)
  __shared__ float sacc;
  if (threadIdx.x == 0) sacc = 0.f;
  __syncthreads();
  float w = wave_reduce_add(c2);
  if ((threadIdx.x & 31) == 0) atomicAdd(&sacc, w);
  __syncthreads();
  if (threadIdx.x == 0) atomicAdd(&ws[C_CNORM], sacc);
}

// ---------------- KB: 11 fused line-search trials (alpha/2^t) ----------------
__global__ __launch_bounds__(256) void kB(float* ws) {
  const int n = blockIdx.x * 256 + threadIdx.x;
  const int l = n & (FRAGLEN - 1);
  const float* ria = ws + RIA_OFF;
  const float* lam = ws + LAM_OFF;

  float ct[11];
  #pragma unroll
  for (int t = 0; t < 11; ++t) ct[t] = 0.f;

  if (l < FRAGLEN - 1) {
    float dx[3], dl[3];
    #pragma unroll
    for (int j = 0; j < 3; ++j) {
      dx[j] = ria[(size_t)(n + 1) * 3 + j] - ria[(size_t)n * 3 + j];
      dl[j] = lam[(size_t)(n + 1) * 3 + j] - lam[(size_t)n * 3 + j];
    }
    float dd[3];                                       // (lam_{i+1}-lam_i) @ M  (v cancels)
    #pragma unroll
    for (int c = 0; c < 3; ++c)
      dd[c] = dl[0] * ws[C_M + 0 + c] + dl[1] * ws[C_M + 3 + c] + dl[2] * ws[C_M + 6 + c];
    float a = ws[C_ALPHA];
    #pragma unroll
    for (int t = 0; t < 11; ++t) {
      float d2 = 0.f;
      #pragma unroll
      for (int c = 0; c < 3; ++c) { float e = dx[c] - a * dd[c]; d2 += e * e; }
      float cc = DISTC * rsqrtf(d2) - 1.f;
      ct[t] = cc * cc;
      a *= 0.5f;
    }
  }

  __shared__ float s[11];
  if (threadIdx.x < 11) s[threadIdx.x] = 0.f;
  __syncthreads();
  #pragma unroll
  for (int t = 0; t < 11; ++t) {
    float w = wave_reduce_add(ct[t]);
    if ((threadIdx.x & 31) == 0) atomicAdd(&s[t], w);
  }
  __syncthreads();
  if (threadIdx.x < 11) atomicAdd(&ws[C_CT + threadIdx.x], s[threadIdx.x]);
}

// ---------------- KC: select alpha, update done/alpha/Asum, reset accumulators
__global__ void kC(float* ws) {
  if (threadIdx.x != 0) return;
  const float cn    = ws[C_CNORM];
  const float alpha = ws[C_ALPHA];
  int   tf = -1;
  float ctv[11];
  for (int t = 0; t < 11; ++t) ctv[t] = ws[C_CT + t];
  for (int t = 0; t < 11; ++t) { if (ctv[t] < cn) { tf = t; break; } }
  float a_f, ctry; int ls;
  if (tf >= 0) { a_f = ldexpf(alpha, -tf); ls = tf; ctry = ctv[tf]; }
  else         { a_f = ldexpf(alpha, -11); ls = 11; ctry = ctv[10]; }
  if (ls == 0 && ctry > CONV) a_f *= 1.5f;             // growth on first-try success
  const float done = ws[C_DONE];
  const float a_apply = (done != 0.f) ? 0.f : a_f;
  if (done == 0.f) ws[C_ALPHA] = a_f;
  ws[C_AAPPLY] = a_apply;
  ws[C_ASUM]  += a_apply;
  if (ctry < CONV) ws[C_DONE] = 1.f;
  ws[C_CNORM] = 0.f;
  for (int t = 0; t < 11; ++t) ws[C_CT + t] = 0.f;
}

// ---------------- KD: L += a*lam3 ; ria -= a*(lam3 @ M + v) ----------------
__global__ __launch_bounds__(256) void kD(float* ws) {
  const int n = blockIdx.x * 256 + threadIdx.x;
  const float a = ws[C_AAPPLY];
  const float* lam = ws + LAM_OFF;
  float* ria = ws + RIA_OFF;
  float* L   = ws + L_OFF;
  float lm[3];
  #pragma unroll
  for (int j = 0; j < 3; ++j) lm[j] = lam[(size_t)n * 3 + j];
  #pragma unroll
  for (int c = 0; c < 3; ++c) {
    float dir = lm[0] * ws[C_M + 0 + c] + lm[1] * ws[C_M + 3 + c]
              + lm[2] * ws[C_M + 6 + c] + ws[C_V + c];
    L[(size_t)n * 3 + c]  += a * lm[c];
    ria[(size_t)n * 3 + c] -= a * dir;
  }
}

// ---------------- K_final: y_out = y0 - L @ Wu_eff.T - Asum*b_u via WMMA
// D = (-L tile 16x4, K=3 padded) x Wu_eff.T(4x16 col-chunk) + C(y0 tile).
__global__ __launch_bounds__(256) void kfinal(const float* __restrict__ y,
                                              const float* __restrict__ bu,
                                              const float* __restrict__ ws,
                                              float* __restrict__ out) {
  const int lane = threadIdx.x & 31;
  const int wave = threadIdx.x >> 5;
  const int tile = blockIdx.x * 8 + wave;    // 16 rows per wave
  const int rb   = tile * 16;
  const int half = lane >> 4;
  const int l16  = lane & 15;

  const float* Lm   = ws + L_OFF;
  const float* wue  = ws + WUE_OFF;
  const float  asum = ws[C_ASUM];

  // A frag: rows of -L (K: lanes 0-15 -> {0,1}, lanes 16-31 -> {2, pad 0})
  v2f a;
  if (half == 0) { a[0] = -Lm[(size_t)(rb + l16) * 3 + 0]; a[1] = -Lm[(size_t)(rb + l16) * 3 + 1]; }
  else           { a[0] = -Lm[(size_t)(rb + l16) * 3 + 2]; a[1] = 0.f; }

  const int rowM0 = rb + half * 8;           // C/D: VGPR r -> row r (lanes 0-15) / r+8
  #pragma unroll
  for (int cc = 0; cc < 8; ++cc) {
    const int col = cc * 16 + l16;
    v2f b;                                   // B[k][n] = Wu_eff[col][k]
    if (half == 0) { b[0] = wue[col * 3 + 0]; b[1] = wue[col * 3 + 1]; }
    else           { b[0] = wue[col * 3 + 2]; b[1] = 0.f; }
    v8f c;
    #pragma unroll
    for (int r = 0; r < 8; ++r) c[r] = y[(size_t)(rowM0 + r) * DY + col];
    v8f d = __builtin_amdgcn_wmma_f32_16x16x4_f32(false, a, false, b,
                                                  (short)0, c, false, false);
    const float bc = asum * bu[col];
    #pragma unroll
    for (int r = 0; r < 8; ++r)
      out[(size_t)(rowM0 + r) * DY + col] = d[r] - bc;
  }
}

extern "C" void kernel_launch(void* const* d_in, const int* in_sizes, int n_in,
                              void* d_out, int out_size, void* d_ws, size_t ws_size,
                              hipStream_t stream) {
  const float* y   = (const float*)d_in[0];
  const float* W_p = (const float*)d_in[1];
  const float* b_p = (const float*)d_in[2];
  const float* W_u = (const float*)d_in[3];
  const float* b_u = (const float*)d_in[4];
  float* out = (float*)d_out;
  float* ws  = (float*)d_ws;   // needs (448 + 3*N3)*4 ≈ 19 MB

  kprep<<<1, 128, 0, stream>>>(W_p, W_u, b_u, ws);
  kproj<<<NPTS / 16 / 8, 256, 0, stream>>>(y, W_p, b_p, ws);
  for (int it = 0; it < NITER; ++it) {
    kA<<<NPTS / 256, 256, 0, stream>>>(ws);
    kB<<<NPTS / 256, 256, 0, stream>>>(ws);
    kC<<<1, 32, 0, stream>>>(ws);
    kD<<<NPTS / 256, 256, 0, stream>>>(ws);
  }
  kfinal<<<NPTS / 16 / 8, 256, 0, stream>>>(y, b_u, ws, out);
}